// SimpleGCN_30331059044547
// MI455X (gfx1250) — compile-verified
//
#include <hip/hip_runtime.h>

// Problem constants (from the reference)
#define Nn    100000   // nodes
#define Ee    800000   // edges
#define IND   300
#define HIDD  250
#define KP    256      // padded K / row stride
#define OUTD  6
#define GG    512      // graphs

typedef float v2f __attribute__((ext_vector_type(2)));
typedef float v8f __attribute__((ext_vector_type(8)));

// Hardware fp32 atomic add, no return (tracked by STOREcnt, RMW at L2).
// Inline asm guarantees global_atomic_add_f32 (no CAS-loop fallback).
__device__ __forceinline__ void atomAddF(float* p, float v) {
  asm volatile("global_atomic_add_f32 %0, %1, off"
               :: "v"(p), "v"(v)
               : "memory");
}

// ---- degree / normalization ------------------------------------------------
__global__ void k_init_deg(float* deg) {
  int i = blockIdx.x * blockDim.x + threadIdx.x;
  if (i < Nn) deg[i] = 1.0f;                       // self-loop
}

__global__ void k_deg_accum(const int* __restrict__ ei, float* deg) {
  int e = blockIdx.x * blockDim.x + threadIdx.x;
  if (e < Ee) atomAddF(&deg[ei[Ee + e]], 1.0f);    // col = ei[1][e]
}

__global__ void k_dinv_s1init(float* deg_dinv, float* s1) {
  int i = blockIdx.x * blockDim.x + threadIdx.x;
  if (i < Nn) {
    float d = rsqrtf(deg_dinv[i]);
    deg_dinv[i] = d;                               // buffer now holds dinv
    s1[i] = d * d;                                 // self-loop norm
  }
}

__global__ void k_s1_accum(const int* __restrict__ ei,
                           const float* __restrict__ dinv, float* s1) {
  int e = blockIdx.x * blockDim.x + threadIdx.x;
  if (e < Ee) {
    int r = ei[e], c = ei[Ee + e];
    atomAddF(&s1[c], dinv[r] * dinv[c]);
  }
}

// ---- h0 = emb[0] @ W1, padded to 256 with zeros ----------------------------
__global__ void k_h0(const float* __restrict__ emb, const float* __restrict__ W1,
                     const float* __restrict__ b1, float* h0p, float* b1p) {
  int k = threadIdx.x;                             // one block of 256
  float h = 0.f, b = 0.f;
  if (k < HIDD) {
    float s = 0.f;
    for (int i = 0; i < IND; ++i) s = fmaf(emb[i], W1[i * HIDD + k], s);
    h = s; b = b1[k];
  }
  h0p[k] = h; b1p[k] = b;
}

__global__ void k_zero_pool(float* pacc, float* cntv) {
  int i = blockIdx.x * blockDim.x + threadIdx.x;
  if (i < GG * KP) pacc[i] = 0.f;
  if (i < GG)      cntv[i] = 0.f;
}

__global__ void k_cnt(const int* __restrict__ batch, float* cntv) {
  int i = blockIdx.x * blockDim.x + threadIdx.x;
  if (i < Nn) atomAddF(&cntv[batch[i]], 1.0f);
}

// ---- z = relu(s1*h0 + b1) @ W2  via V_WMMA_F32_16X16X4_F32 -----------------
// A tile regenerated in-register from scalar s1[row] (never materialize x1).
__global__ __launch_bounds__(256) void k_gemm(const float* __restrict__ s1,
                                              const float* __restrict__ h0p,
                                              const float* __restrict__ b1p,
                                              const float* __restrict__ W2,
                                              float* __restrict__ z) {
  const int wave = threadIdx.x >> 5;
  const int lane = threadIdx.x & 31;
  const int rt   = blockIdx.x * 8 + wave;          // 16-row tile id
  if (rt >= Nn / 16) return;                       // wave-uniform exit
  const int half  = lane >> 4;
  const int l15   = lane & 15;
  const int Mbase = rt * 16;
  const float s1v = s1[Mbase + l15];

  for (int cp = 0; cp < 4; ++cp) {                 // 4 passes x 64 columns
    v8f acc[4] = {v8f{}, v8f{}, v8f{}, v8f{}};
    const int nb = cp * 64;
    for (int ks = 0; ks < 64; ++ks) {              // K = 256 (padded)
      const int k0 = ks * 4 + 2 * half;            // A/B ISA lane layout
      const int k1 = k0 + 1;
      // A fragment on the fly; k>=250 gives exact 0 (h0p/b1p zero-padded)
      v2f a;
      a.x = fmaxf(fmaf(s1v, h0p[k0], b1p[k0]), 0.f);
      a.y = fmaxf(fmaf(s1v, h0p[k1], b1p[k1]), 0.f);
      const int kc0 = k0 < HIDD ? k0 : HIDD - 1;   // clamp: a==0 there anyway
      const int kc1 = k1 < HIDD ? k1 : HIDD - 1;
#pragma unroll
      for (int t = 0; t < 4; ++t) {
        int n  = nb + t * 16 + l15;
        int nc = n < HIDD ? n : HIDD - 1;          // padded cols never stored
        v2f b;
        b.x = W2[kc0 * HIDD + nc];
        b.y = W2[kc1 * HIDD + nc];
        acc[t] = __builtin_amdgcn_wmma_f32_16x16x4_f32(
            false, a, false, b, (short)0, acc[t], false, false);
      }
    }
#pragma unroll
    for (int t = 0; t < 4; ++t) {
      int n = nb + t * 16 + l15;
      if (n < HIDD) {
#pragma unroll
        for (int r = 0; r < 8; ++r)                // C/D: M = r + 8*half
          z[(size_t)(Mbase + r + 8 * half) * KP + n] = acc[t][r];
      }
    }
  }
}

// ---- edge propagation fused with pooling: pacc[g] += norm * z[row] ---------
__global__ __launch_bounds__(256) void k_prop(const int* __restrict__ ei,
                                              const int* __restrict__ batch,
                                              const float* __restrict__ dinv,
                                              const float* __restrict__ z,
                                              float* pacc) {
  int wave = threadIdx.x >> 5;
  int lane = threadIdx.x & 31;
  int e = blockIdx.x * 8 + wave;                   // one wave per edge
  if (e >= Ee + Nn) return;
  int r, c;
  if (e < Ee) { r = ei[e]; c = ei[Ee + e]; }
  else        { r = c = e - Ee; }                  // self-loops
  float norm = dinv[r] * dinv[c];
  int g = batch[c];
  const float* zr = z + (size_t)r * KP;
  float* pg = pacc + (size_t)g * KP;
  for (int k = lane; k < HIDD; k += 32)
    atomAddF(&pg[k], norm * zr[k]);
}

// ---- out[g,o] = ((pacc[g]+cnt*b2)/max(cnt,1)) @ fcW + fcb ------------------
__global__ void k_out(const float* __restrict__ pacc, const float* __restrict__ cntv,
                      const float* __restrict__ b2, const float* __restrict__ fcW,
                      const float* __restrict__ fcb, float* out) {
  int idx = blockIdx.x * blockDim.x + threadIdx.x;
  if (idx >= GG * OUTD) return;
  int g = idx / OUTD, o = idx % OUTD;
  float c   = cntv[g];
  float inv = 1.0f / fmaxf(c, 1.0f);
  const float* pg = pacc + (size_t)g * KP;
  float s = 0.f;
  for (int k = 0; k < HIDD; ++k) {
    float p = (pg[k] + c * b2[k]) * inv;
    s = fmaf(p, fcW[k * OUTD + o], s);
  }
  out[idx] = s + fcb[o];
}

extern "C" void kernel_launch(void* const* d_in, const int* in_sizes, int n_in,
                              void* d_out, int out_size, void* d_ws, size_t ws_size,
                              hipStream_t stream) {
  const int*   ei    = (const int*)d_in[1];   // [2,E]
  const int*   batch = (const int*)d_in[2];   // [N]
  const float* emb   = (const float*)d_in[4]; // [1,300]
  const float* W1    = (const float*)d_in[5]; // [300,250]
  const float* b1    = (const float*)d_in[6]; // [250]
  const float* W2    = (const float*)d_in[7]; // [250,250]
  const float* b2    = (const float*)d_in[8]; // [250]
  const float* fcW   = (const float*)d_in[9]; // [250,6]
  const float* fcb   = (const float*)d_in[10];// [6]
  float* out = (float*)d_out;

  // Workspace layout (floats): ~103.7 MB total
  float* ws   = (float*)d_ws;
  float* deg  = ws;                       // [N]  -> becomes dinv in-place
  float* s1   = ws + Nn;                  // [N]
  float* h0p  = ws + 2 * (size_t)Nn;      // [256] zero-padded
  float* b1p  = h0p + KP;                 // [256] zero-padded
  float* pacc = b1p + KP;                 // [512*256]
  float* cntv = pacc + (size_t)GG * KP;   // [512]
  float* z    = cntv + GG;                // [N*256]

  k_init_deg   <<<(Nn + 255) / 256, 256, 0, stream>>>(deg);
  k_deg_accum  <<<(Ee + 255) / 256, 256, 0, stream>>>(ei, deg);
  k_dinv_s1init<<<(Nn + 255) / 256, 256, 0, stream>>>(deg, s1);
  k_s1_accum   <<<(Ee + 255) / 256, 256, 0, stream>>>(ei, deg, s1); // deg==dinv now
  k_h0         <<<1, 256, 0, stream>>>(emb, W1, b1, h0p, b1p);
  k_zero_pool  <<<(GG * KP + 255) / 256, 256, 0, stream>>>(pacc, cntv);
  k_cnt        <<<(Nn + 255) / 256, 256, 0, stream>>>(batch, cntv);
  k_gemm       <<<(Nn / 16 + 7) / 8, 256, 0, stream>>>(s1, h0p, b1p, W2, z);
  k_prop       <<<(Ee + Nn + 7) / 8, 256, 0, stream>>>(ei, batch, deg, z, pacc);
  k_out        <<<(GG * OUTD + 255) / 256, 256, 0, stream>>>(pacc, cntv, b2, fcW, fcb, out);
}